// KPbyNode_51007031607813
// MI455X (gfx1250) — compile-verified
//
#include <hip/hip_runtime.h>
#include <hip/hip_bf16.h>
#include <math.h>

#define Nn 200
#define Pp 200
#define Tt 9000
#define Ss 360
#define STRIDE 25
#define Bb 32
#define Hh 128
#define TOPK 3

typedef float v2f __attribute__((ext_vector_type(2)));
typedef float v8f __attribute__((ext_vector_type(8)));
typedef __attribute__((address_space(1))) int as1_int;
typedef __attribute__((address_space(3))) int as3_int;

// ---------------- zero graphs region + accumulators ----------------
__global__ void zero_kernel(float* graphs, float* acc, int n) {
    int i = blockIdx.x * blockDim.x + threadIdx.x;
    if (i < n) graphs[i] = 0.0f;
    if (i < 2) acc[i] = 0.0f;
}

// ---------------- pattern prep: np_, top-3, tp, rp, norms ----------------
__global__ void prep_patterns(const float* __restrict__ patterns,
                              float* __restrict__ npat, int* __restrict__ topidx,
                              float* __restrict__ tpval, float* __restrict__ rpval,
                              float* __restrict__ acc) {
    const int p = blockIdx.x;
    const int tid = threadIdx.x;                // 256 threads
    __shared__ float red[256];
    __shared__ int   ri[256];
    __shared__ float uval[256];
    __shared__ float av[256];

    float v = 0.0f;
    if (tid < Nn) {
        float pv = patterns[p * Nn + tid];
        v = (tid == p) ? 1.0f : tanhf(pv);      // tanh(pat*(1-I)) + I
    }
    red[tid] = v * v;
    __syncthreads();
    for (int off = 128; off; off >>= 1) {
        if (tid < off) red[tid] += red[tid + off];
        __syncthreads();
    }
    const float s2 = red[0];
    const float inv = 1.0f / sqrtf(s2 + 1e-9f);
    if (tid == 0) atomicAdd(&acc[1], s2 * inv * inv);   // ||np_||_F^2 contribution
    __syncthreads();

    const float u = v * inv;
    if (tid < Nn) npat[p * Nn + tid] = u;
    uval[tid] = (tid < Nn) ? u : 0.0f;
    av[tid]   = (tid < Nn) ? fabsf(u) : -1.0f;
    __syncthreads();

    int wins[TOPK];
    for (int k = 0; k < TOPK; ++k) {
        red[tid] = av[tid]; ri[tid] = tid;
        __syncthreads();
        for (int off = 128; off; off >>= 1) {
            if (tid < off) {
                float v2 = red[tid + off]; int i2 = ri[tid + off];
                if (v2 > red[tid] || (v2 == red[tid] && i2 < ri[tid])) {
                    red[tid] = v2; ri[tid] = i2;
                }
            }
            __syncthreads();
        }
        wins[k] = ri[0];
        __syncthreads();
        if (tid == 0) av[wins[k]] = -2.0f;      // mask out for next pass
        __syncthreads();
    }
    if (tid == 0) {
        float t2 = 0.0f;
        for (int k = 0; k < TOPK; ++k) {
            int w = wins[k];
            float tv = uval[w];
            topidx[p * TOPK + k] = w;
            tpval[p * TOPK + k]  = tv;
            rpval[p * TOPK + k]  = ((tv > 0.0f) ? 1.0f : (tv < 0.0f) ? -1.0f : 0.0f) / (float)TOPK;
            t2 += tv * tv;
        }
        atomicAdd(&acc[0], t2);                 // ||tp||_F^2
    }
}

// ---------------- per-frame inverse norms ----------------
__global__ void frame_norms(const float* __restrict__ x, float* __restrict__ rn) {
    int i = blockIdx.x * blockDim.x + threadIdx.x;
    if (i >= Bb * Tt) return;
    int b = i / Tt, t = i % Tt;
    const float* xb = x + (size_t)b * Nn * Tt + t;
    float s = 0.0f;
    for (int n = 0; n < Nn; ++n) { float v = xb[(size_t)n * Tt]; s += v * v; }
    rn[i] = 1.0f / sqrtf(s + 1e-9f);
}

// ---------------- WMMA fp32 GEMM: scores[t,p] for one batch ----------------
// A (16t x 4k) = nf tile (global, clamped+masked), B (4k x 16p) = np_ tile
// staged in LDS via CDNA5 async global->LDS copy, read back as ds_load_2addr_b64.
__global__ void __launch_bounds__(256)
gemm_scores(const float* __restrict__ x, const float* __restrict__ rn,
            const float* __restrict__ npat, float* __restrict__ scores, int b) {
    __shared__ float bpat[16 * Nn];             // 12.8 KB: 16 patterns x 200 k
    const int p0 = blockIdx.y * 16;
    const int nvalid = (Pp - p0 < 16) ? (Pp - p0) : 16;

    // zero-fill rows past P (last p-tile), then async-stage the valid rows
    for (int i = threadIdx.x + nvalid * Nn; i < 16 * Nn; i += 256) bpat[i] = 0.0f;
#if __has_builtin(__builtin_amdgcn_global_load_async_to_lds_b32)
    {
        as1_int* gp = (as1_int*)(npat + (size_t)p0 * Nn);
        as3_int* lp = (as3_int*)bpat;
        for (int i = threadIdx.x; i < nvalid * Nn; i += 256)
            __builtin_amdgcn_global_load_async_to_lds_b32(gp + i, lp + i, 0, 0);
#if __has_builtin(__builtin_amdgcn_s_wait_asynccnt)
        __builtin_amdgcn_s_wait_asynccnt(0);
#else
        asm volatile("s_wait_asynccnt 0x0" ::: "memory");
#endif
    }
#else
    for (int i = threadIdx.x; i < nvalid * Nn; i += 256)
        bpat[i] = npat[(size_t)p0 * Nn + i];
#endif
    __syncthreads();                            // before any wave-uniform exit

    const int wave = threadIdx.x >> 5;
    const int lane = threadIdx.x & 31;
    const int t0 = (blockIdx.x * 8 + wave) * 16;
    if (t0 >= Tt) return;                       // wave-uniform: EXEC all-1 at WMMA
    const int row = lane & 15;
    const int hi  = lane >> 4;

    const int tt  = t0 + row;
    const bool tv = (tt < Tt);
    const int tta = tv ? tt : (Tt - 1);         // clamped address, masked by rr
    const float rr = rn[b * Tt + tta] * (tv ? 1.0f : 0.0f);
    const float* xb = x + (size_t)b * Nn * Tt + tta;
    const float* bp = bpat + row * Nn;          // row*200 even -> 8B aligned v2f reads

    v8f c = {};
#pragma unroll 2
    for (int k = 0; k < Nn; k += 4) {
        const int ka = k + hi * 2;              // lanes 16-31 hold K=2,3 (ISA A-layout)
        v2f a, bm;
        a.x = xb[(size_t)ka * Tt] * rr;         // unconditional loads, no EXEC churn
        a.y = xb[(size_t)(ka + 1) * Tt] * rr;
        bm  = *(const v2f*)(bp + ka);           // ds_load b64 (pairs fuse to 2addr)
        c = __builtin_amdgcn_wmma_f32_16x16x4_f32(false, a, false, bm,
                                                  (short)0, c, false, false);
    }

    // epilogue: C layout -> scores[t][p]; fast path for fully-interior tiles
    const int pst = p0 + row;
    float* sb = scores + (size_t)(t0 + hi * 8) * Pp + pst;   // row base for this lane
    if (t0 + 15 < Tt && p0 + 15 < Pp) {         // wave-uniform: no EXEC churn
#pragma unroll
        for (int j = 0; j < 8; ++j) sb[(size_t)j * Pp] = c[j];
    } else {
#pragma unroll
        for (int j = 0; j < 8; ++j) {
            int t = t0 + j + hi * 8;
            if (t < Tt && pst < Pp) sb[(size_t)j * Pp] = c[j];
        }
    }
}

// ---------------- max/min pooling with argmax indices ----------------
__global__ void pool_kernel(const float* __restrict__ scores,
                            int* __restrict__ idxp, int* __restrict__ idxn) {
    int s = blockIdx.x;
    int p = threadIdx.x;
    if (p >= Pp) return;
    const float* base = scores + (size_t)(s * STRIDE) * Pp + p;
    float mx = base[0], mn = base[0];
    int mxi = 0, mni = 0;
    for (int i = 1; i < STRIDE; ++i) {
        float v = base[(size_t)i * Pp];
        if (v > mx) { mx = v; mxi = i; }        // first-occurrence argmax
        if (v < mn) { mn = v; mni = i; }        // first-occurrence argmax(-v)
    }
    idxp[s * Pp + p] = s * STRIDE + mxi;
    idxn[s * Pp + p] = s * STRIDE + mni;
}

// ---------------- per-(b,p): finalscore + masked avg_sel -> diag ----------------
__global__ void __launch_bounds__(512)
finalize_bp(const float* __restrict__ x, const float* __restrict__ rn,
            const float* __restrict__ scores, const int* __restrict__ idxp,
            const int* __restrict__ idxn, const int* __restrict__ topidx,
            const float* __restrict__ tpval, float* __restrict__ diag, int b) {
    const int p = blockIdx.x;
    const int s = threadIdx.x;                  // 512 threads, 360 active
    const int n0 = topidx[p * TOPK + 0];
    const int n1 = topidx[p * TOPK + 1];
    const int n2 = topidx[p * TOPK + 2];
    float a0 = 0.0f, a1 = 0.0f, a2 = 0.0f, a3 = 0.0f;
    if (s < Ss) {
        int ip = idxp[s * Pp + p];
        int im = idxn[s * Pp + p];
        a0 = scores[(size_t)ip * Pp + p] - scores[(size_t)im * Pp + p]; // possum + negsum
        float rp_ = rn[b * Tt + ip];
        float rm_ = rn[b * Tt + im];
        const float* xb = x + (size_t)b * Nn * Tt;
        a1 = xb[(size_t)n0 * Tt + ip] * rp_ - xb[(size_t)n0 * Tt + im] * rm_;
        a2 = xb[(size_t)n1 * Tt + ip] * rp_ - xb[(size_t)n1 * Tt + im] * rm_;
        a3 = xb[(size_t)n2 * Tt + ip] * rp_ - xb[(size_t)n2 * Tt + im] * rm_;
    }
    __shared__ float r0[512], r1[512], r2[512], r3[512];
    r0[s] = a0; r1[s] = a1; r2[s] = a2; r3[s] = a3;
    __syncthreads();
    for (int off = 256; off; off >>= 1) {
        if (s < off) {
            r0[s] += r0[s + off]; r1[s] += r1[s + off];
            r2[s] += r2[s + off]; r3[s] += r3[s + off];
        }
        __syncthreads();
    }
    if (s == 0) {
        const float Sf = (float)Ss;
        float fsc = r0[0] / Sf * 0.5f;                      // (compos+comneg)/2
        float cc = sqrtf(200.0f / (float)TOPK) / Sf * 0.5f; // avg_sel scale
        float v0 = cc * r1[0], v1 = cc * r2[0], v2 = cc * r3[0];
        float t0 = tpval[p * TOPK + 0], t1 = tpval[p * TOPK + 1], t2 = tpval[p * TOPK + 2];
        float dot = v0 * t0 + v1 * t1 + v2 * t2;
        float nT  = sqrtf(v0 * v0 + v1 * v1 + v2 * v2 + 1e-9f);
        float ntp = sqrtf(t0 * t0 + t1 * t1 + t2 * t2 + 1e-9f);
        float sc2 = dot / (nT * ntp);
        diag[b * Pp + p] = fsc + 0.25f * sc2 * sqrtf((float)TOPK / 200.0f);
    }
}

// ---------------- graphs: sparse rank-1 scatter (3x3 per (b,p)) ----------------
__global__ void graphs_scatter(const float* __restrict__ diag,
                               const int* __restrict__ topidx,
                               const float* __restrict__ rpval,
                               float* __restrict__ graphs) {
    int idx = blockIdx.x * blockDim.x + threadIdx.x;
    if (idx >= Bb * Pp) return;
    int b = idx / Pp, p = idx % Pp;
    float d = diag[idx];
    int   ni[TOPK]; float rv[TOPK];
#pragma unroll
    for (int k = 0; k < TOPK; ++k) { ni[k] = topidx[p * TOPK + k]; rv[k] = rpval[p * TOPK + k]; }
#pragma unroll
    for (int i = 0; i < TOPK; ++i)
#pragma unroll
        for (int j = 0; j < TOPK; ++j) {
            // non_diag (3 off / 1 diag) divided by K/5 -> x5 off-diag, x5/3 diag
            float f = (ni[i] == ni[j]) ? (5.0f / 3.0f) : 5.0f;
            atomicAdd(&graphs[((size_t)b * Nn + ni[i]) * Nn + ni[j]], d * rv[i] * rv[j] * f);
        }
}

// ---------------- classifier: relu(graphs@W1+b1), mean, @W2+b2 ----------------
__global__ void __launch_bounds__(128)
classifier(const float* __restrict__ graphs, const float* __restrict__ W1,
           const float* __restrict__ b1, const float* __restrict__ W2,
           const float* __restrict__ b2, float* __restrict__ out) {
    const int b = blockIdx.x;
    const int hh = threadIdx.x;                 // 128 threads = H
    __shared__ float row[Nn];
    __shared__ float mh[Hh];
    const float bias = b1[hh];
    float acc = 0.0f;
    for (int n = 0; n < Nn; ++n) {
        __syncthreads();
        for (int m = hh; m < Nn; m += Hh) row[m] = graphs[((size_t)b * Nn + n) * Nn + m];
        __syncthreads();
        float sv = bias;
        for (int m = 0; m < Nn; ++m) sv += row[m] * W1[m * Hh + hh];
        acc += fmaxf(sv, 0.0f);
    }
    mh[hh] = acc / (float)Nn;
    __syncthreads();
    if (hh < 2) {
        float o = b2[hh];
        for (int h = 0; h < Hh; ++h) o += mh[h] * W2[h * 2 + hh];
        out[b * 2 + hh] = o;
    }
}

// ---------------- norm scalar ----------------
__global__ void finish_norm(const float* __restrict__ acc, float* __restrict__ nout) {
    if (threadIdx.x == 0 && blockIdx.x == 0) {
        float r = 1.0f - sqrtf(acc[0]) / sqrtf(acc[1]);
        nout[0] = r * r;
    }
}

static inline size_t align256(size_t x) { return (x + 255) & ~(size_t)255; }

extern "C" void kernel_launch(void* const* d_in, const int* in_sizes, int n_in,
                              void* d_out, int out_size, void* d_ws, size_t ws_size,
                              hipStream_t stream) {
    const float* x        = (const float*)d_in[0];
    // d_in[1] = length (static 9000, unused)
    const float* patterns = (const float*)d_in[2];
    const float* W1       = (const float*)d_in[3];
    const float* b1       = (const float*)d_in[4];
    const float* W2       = (const float*)d_in[5];
    const float* b2       = (const float*)d_in[6];

    float* out      = (float*)d_out;          // [B,2] = 64 floats
    float* norm_out = out + Bb * 2;           // scalar at [64]
    float* graphs   = out + Bb * 2 + 1;       // [B,N,N] at [65..)

    // workspace carve-up (~9 MB)
    char* wsb = (char*)d_ws;
    size_t off = 0;
    float* w_np   = (float*)(wsb + off); off = align256(off + sizeof(float) * Pp * Nn);
    int*   w_tidx = (int*)  (wsb + off); off = align256(off + sizeof(int)   * Pp * TOPK);
    float* w_tp   = (float*)(wsb + off); off = align256(off + sizeof(float) * Pp * TOPK);
    float* w_rp   = (float*)(wsb + off); off = align256(off + sizeof(float) * Pp * TOPK);
    float* w_acc  = (float*)(wsb + off); off = align256(off + sizeof(float) * 2);
    float* w_rn   = (float*)(wsb + off); off = align256(off + sizeof(float) * Bb * Tt);
    float* w_sc   = (float*)(wsb + off); off = align256(off + sizeof(float) * (size_t)Tt * Pp);
    int*   w_ip   = (int*)  (wsb + off); off = align256(off + sizeof(int)   * Ss * Pp);
    int*   w_in   = (int*)  (wsb + off); off = align256(off + sizeof(int)   * Ss * Pp);
    float* w_diag = (float*)(wsb + off); off = align256(off + sizeof(float) * Bb * Pp);
    (void)ws_size; (void)in_sizes; (void)n_in; (void)out_size;

    const int ngraph = Bb * Nn * Nn;
    zero_kernel<<<(ngraph + 255) / 256, 256, 0, stream>>>(graphs, w_acc, ngraph);
    prep_patterns<<<Pp, 256, 0, stream>>>(patterns, w_np, w_tidx, w_tp, w_rp, w_acc);
    frame_norms<<<(Bb * Tt + 255) / 256, 256, 0, stream>>>(x, w_rn);

    const int nttiles = (Tt + 15) / 16;               // 563
    dim3 ggrid((nttiles + 7) / 8, (Pp + 15) / 16);    // (71, 13)
    for (int b = 0; b < Bb; ++b) {
        gemm_scores<<<ggrid, 256, 0, stream>>>(x, w_rn, w_np, w_sc, b);
        pool_kernel<<<Ss, 256, 0, stream>>>(w_sc, w_ip, w_in);
        finalize_bp<<<Pp, 512, 0, stream>>>(x, w_rn, w_sc, w_ip, w_in,
                                            w_tidx, w_tp, w_diag, b);
    }

    graphs_scatter<<<(Bb * Pp + 255) / 256, 256, 0, stream>>>(w_diag, w_tidx, w_rp, graphs);
    classifier<<<Bb, 128, 0, stream>>>(graphs, W1, b1, W2, b2, out);
    finish_norm<<<1, 1, 0, stream>>>(w_acc, norm_out);
}